// retina_polar_15728170238139
// MI455X (gfx1250) — compile-verified
//
#include <hip/hip_runtime.h>
#include <hip/hip_bf16.h>
#include <stdint.h>

// Problem constants (match reference)
#define B_    64
#define C_    3
#define HI_   224
#define WI_   224
#define HP_   32      // pooled H
#define WP_   64      // pooled W
#define UR_   10
#define UT_   10
#define WG_   (WP_*UT_)          // 640 angles
#define BPB   4                  // batches per block (grid-slab reuse factor)
#define THREADS (BPB*WP_)        // 256 threads = 8 waves (wave32)

#define SLAB_FLOATS (UR_*WG_*2)  // 12800 floats = 51.2 KB
#define SLAB_BYTES  (SLAB_FLOATS*4)
#define SLAB_XFERS  (SLAB_BYTES/16)   // 3200 b128 lane-transfers

__global__ __launch_bounds__(THREADS)
void retina_polar_kernel(const float* __restrict__ x,
                         const float* __restrict__ lt,
                         const float* __restrict__ grid2d,
                         float* __restrict__ out)
{
    // grid slab for pooled row h: rows h*10 .. h*10+9, all 640 angles, (x,y)
    __shared__ __align__(16) float s_grid[SLAB_FLOATS];

    const int h  = blockIdx.x;            // pooled row 0..31
    const int bg = blockIdx.y;            // batch group 0..15
    const int t  = threadIdx.x;
    const int bl = t >> 6;                // batch-in-block 0..3
    const int w  = t & 63;                // output angle 0..63
    const int b  = bg * BPB + bl;

    // ---- CDNA5 async bulk copy: global grid slab -> LDS (ASYNCcnt path) ----
    {
        const uint64_t src  = (uint64_t)(uintptr_t)grid2d + (uint64_t)h * SLAB_BYTES;
        // generic->LDS: low 32 bits of the flat LDS-aperture address are the LDS byte offset
        const uint32_t lds0 = (uint32_t)(uintptr_t)s_grid;
        for (int idx = t; idx < SLAB_XFERS; idx += THREADS) {
            uint32_t voff = (uint32_t)idx * 16u;       // byte offset into slab
            uint32_t ldsa = lds0 + voff;               // per-lane LDS dest
            asm volatile("global_load_async_to_lds_b128 %0, %1, %2"
                         :: "v"(ldsa), "v"(voff), "s"(src)
                         : "memory");
        }
        asm volatile("s_wait_asynccnt 0x0" ::: "memory");
    }
    __syncthreads();

    // Glimpse center in pixel space: ix = ((gx+1)*224-1)*0.5 = gx*112 + 111.5
    const float cx = fmaf(lt[b*2 + 0], 112.0f, 111.5f);
    const float cy = fmaf(lt[b*2 + 1], 112.0f, 111.5f);

    const float* __restrict__ p0 = x + (size_t)b * (C_*HI_*WI_);
    const float* __restrict__ p1 = p0 + HI_*WI_;
    const float* __restrict__ p2 = p1 + HI_*WI_;

    float acc0 = 0.f, acc1 = 0.f, acc2 = 0.f;

    const float2* __restrict__ sg = (const float2*)s_grid;

    #pragma unroll 1
    for (int ur = 0; ur < UR_; ++ur) {
        const float2* __restrict__ row = sg + ur * WG_ + w * UT_;
        #pragma unroll 2
        for (int ut = 0; ut < UT_; ++ut) {
            float2 g  = row[ut];                       // ds_load_b64
            float ix  = fmaf(g.x, 112.0f, cx);
            float iy  = fmaf(g.y, 112.0f, cy);
            float x0f = floorf(ix);
            float y0f = floorf(iy);
            float fx  = ix - x0f;
            float fy  = iy - y0f;
            int   x0  = (int)x0f;
            int   y0  = (int)y0f;

            // per-corner validity (padding_mode='zeros'), folded into weights
            bool vx0 = (x0 >= 0)  & (x0 < WI_);
            bool vx1 = (x0 >= -1) & (x0 < WI_ - 1);
            bool vy0 = (y0 >= 0)  & (y0 < HI_);
            bool vy1 = (y0 >= -1) & (y0 < HI_ - 1);

            float gx1 = fx,        gx0 = 1.0f - fx;
            float gy1 = fy,        gy0 = 1.0f - fy;
            float w00 = (vx0 & vy0) ? gx0 * gy0 : 0.0f;
            float w01 = (vx1 & vy0) ? gx1 * gy0 : 0.0f;
            float w10 = (vx0 & vy1) ? gx0 * gy1 : 0.0f;
            float w11 = (vx1 & vy1) ? gx1 * gy1 : 0.0f;

            // clamped addresses -> loads always in-bounds, unconditional
            int xc0 = min(max(x0,     0), WI_ - 1);
            int xc1 = min(max(x0 + 1, 0), WI_ - 1);
            int yc0 = min(max(y0,     0), HI_ - 1);
            int yc1 = min(max(y0 + 1, 0), HI_ - 1);

            int o00 = yc0 * WI_ + xc0;
            int o01 = yc0 * WI_ + xc1;
            int o10 = yc1 * WI_ + xc0;
            int o11 = yc1 * WI_ + xc1;

            acc0 += w00*p0[o00] + w01*p0[o01] + w10*p0[o10] + w11*p0[o11];
            acc1 += w00*p1[o00] + w01*p1[o01] + w10*p1[o10] + w11*p1[o11];
            acc2 += w00*p2[o00] + w01*p2[o01] + w10*p2[o10] + w11*p2[o11];
        }
    }

    const float s = 1.0f / (float)(UR_ * UT_);
    out[((size_t)(b*C_ + 0)*HP_ + h)*WP_ + w] = acc0 * s;
    out[((size_t)(b*C_ + 1)*HP_ + h)*WP_ + w] = acc1 * s;
    out[((size_t)(b*C_ + 2)*HP_ + h)*WP_ + w] = acc2 * s;
}

extern "C" void kernel_launch(void* const* d_in, const int* in_sizes, int n_in,
                              void* d_out, int out_size, void* d_ws, size_t ws_size,
                              hipStream_t stream) {
    const float* x      = (const float*)d_in[0];   // [64,3,224,224]
    const float* lt     = (const float*)d_in[1];   // [64,2]
    const float* grid2d = (const float*)d_in[2];   // [320,640,2]
    float* out          = (float*)d_out;           // [64,3,32,64]

    dim3 grid(HP_, B_ / BPB);   // (32, 16)
    dim3 block(THREADS);        // 256
    hipLaunchKernelGGL(retina_polar_kernel, grid, block, 0, stream,
                       x, lt, grid2d, out);
}